// CRF_46995532153476
// MI455X (gfx1250) — compile-verified
//
#include <hip/hip_runtime.h>
#include <hip/hip_bf16.h>

// Batched Viterbi (max-sum) decode, fused: per-block WMMA emissions -> LDS,
// then in-LDS recursion + backtrack. One block per batch element.
//
// Sizes fixed by the reference: B=1024, S=256, D=128, K=26.

#define BB 1024
#define SS 256
#define DD 128
#define KK 26

typedef __attribute__((ext_vector_type(2))) float v2f;
typedef __attribute__((ext_vector_type(8))) float v8f;

__global__ __launch_bounds__(256) void viterbi_fused_kernel(
    const float* __restrict__ X,   // [B,S,D]
    const float* __restrict__ W,   // [K,D]
    const float* __restrict__ T,   // [K,K]  T[kp][kn]
    int* __restrict__ out)         // [B,S] int32 labels
{
    __shared__ float         emis[SS * KK];      // 26624 B
    __shared__ unsigned char bp[SS * KK];        //  6656 B (t=1..S-1 used)
    __shared__ int           path[SS];           //  1024 B

    const int b    = blockIdx.x;
    const int tid  = threadIdx.x;
    const int lane = tid & 31;
    const int wave = tid >> 5;

    const float* __restrict__ Xb = X + (size_t)b * SS * DD;

    // ---------------- Phase 1: emissions via V_WMMA_F32_16X16X4_F32 ----------
    // A (16x4, MxK): lanes 0-15 hold M=lane, K={0,1}; lanes 16-31 hold K={2,3}.
    // B (4x16, KxN): lanes 0-15 hold N=lane, K={0,1}; lanes 16-31 hold K={2,3}.
    // C (16x16):     VGPR r holds M = r + 8*(lane>=16), N = lane&15.
    const int  half = lane >> 4;      // 0 or 1
    const int  l15  = lane & 15;
    const int  koff = half * 2;       // K-offset inside the 4-wide chunk

    const int  kb0  = l15;            // N-tile 0: label column 0..15
    const int  kb1  = 16 + l15;       // N-tile 1: label column 16..31
    const bool kb1v = (kb1 < KK);
    // Out-of-range label columns: clamp the W row index and load real data
    // unconditionally. B column n only affects C column n, and those C columns
    // are never stored, so no zeroing / predication is needed in the hot loop.
    const int  kb1c = kb1v ? kb1 : (KK - 1);

    const float* __restrict__ wrow0 = W + (size_t)kb0  * DD + koff;
    const float* __restrict__ wrow1 = W + (size_t)kb1c * DD + koff;

    // 16 M-tiles of t, 8 waves -> 2 tiles per wave.
    for (int mt = wave * 2; mt < wave * 2 + 2; ++mt) {
        const int t0 = mt * 16;
        const float* __restrict__ arow = Xb + (size_t)(t0 + l15) * DD + koff;

        v8f acc0 = {};
        v8f acc1 = {};
        for (int d0 = 0; d0 < DD; d0 += 4) {
            v2f a;
            a.x = arow[d0];
            a.y = arow[d0 + 1];
            v2f b0;
            b0.x = wrow0[d0];
            b0.y = wrow0[d0 + 1];
            v2f b1;
            b1.x = wrow1[d0];
            b1.y = wrow1[d0 + 1];
            acc0 = __builtin_amdgcn_wmma_f32_16x16x4_f32(
                false, a, false, b0, (short)0, acc0, false, false);
            acc1 = __builtin_amdgcn_wmma_f32_16x16x4_f32(
                false, a, false, b1, (short)0, acc1, false, false);
        }

        // Scatter C tiles into LDS emis[t][k].
        #pragma unroll
        for (int r = 0; r < 8; ++r) {
            const int t = t0 + r + 8 * half;
            emis[t * KK + l15] = acc0[r];
            if (kb1v) emis[t * KK + 16 + l15] = acc1[r];
        }
    }
    __syncthreads();

    // ---------------- Phase 2: Viterbi recursion (wave 0 only) ---------------
    if (wave == 0) {
        // Lane kn holds column T[:, kn] in registers (fully unrolled).
        const int kn = (lane < KK) ? lane : 0;
        float tcol[KK];
        #pragma unroll
        for (int kp = 0; kp < KK; ++kp) tcol[kp] = T[kp * KK + kn];

        const float NEG_INF = -__builtin_inff();
        float delta = (lane < KK) ? emis[lane] : NEG_INF;

        for (int t = 1; t < SS; ++t) {
            float best = NEG_INF;
            int   arg  = 0;
            #pragma unroll
            for (int kp = 0; kp < KK; ++kp) {
                const float dkp = __shfl(delta, kp, 32);
                const float v   = dkp + tcol[kp];
                if (v > best) { best = v; arg = kp; }   // strict > keeps lowest kp on ties
            }
            const float e = (lane < KK) ? emis[t * KK + lane] : NEG_INF;
            delta = best + e;                           // lanes >=26 stay -inf
            if (lane < KK) bp[t * KK + lane] = (unsigned char)arg;
        }

        // argmax over final delta, lowest index wins ties.
        float bv = delta;
        int   bi = (lane < KK) ? lane : 0x7fffffff;
        #pragma unroll
        for (int off = 16; off > 0; off >>= 1) {
            const float ov = __shfl_down(bv, off, 32);
            const int   oi = __shfl_down(bi, off, 32);
            if (ov > bv || (ov == bv && oi < bi)) { bv = ov; bi = oi; }
        }
        // Backtrack on lane 0 (bp written earlier by this same wave: DS ops
        // from one wave complete in order, no barrier needed).
        if (lane == 0) {
            int cur = bi;
            path[SS - 1] = cur;
            for (int t = SS - 1; t >= 1; --t) {
                cur = (int)bp[t * KK + cur];
                path[t - 1] = cur;
            }
        }
    }
    __syncthreads();

    // ---------------- Phase 3: coalesced path write-out ----------------------
    out[(size_t)b * SS + tid] = path[tid];
}

extern "C" void kernel_launch(void* const* d_in, const int* in_sizes, int n_in,
                              void* d_out, int out_size, void* d_ws, size_t ws_size,
                              hipStream_t stream) {
    const float* X = (const float*)d_in[0];   // [B,S,D] fp32
    const float* W = (const float*)d_in[1];   // [K,D]   fp32
    const float* T = (const float*)d_in[2];   // [K,K]   fp32
    int* out = (int*)d_out;                   // [B,S]   int32

    const int B = in_sizes[0] / (SS * DD);    // = 1024
    viterbi_fused_kernel<<<dim3(B), dim3(256), 0, stream>>>(X, W, T, out);
}